// ContinuousEmbedding_18700287607510
// MI455X (gfx1250) — compile-verified
//
#include <hip/hip_runtime.h>

typedef __attribute__((ext_vector_type(2))) float v2f;
typedef __attribute__((ext_vector_type(4))) float v4f;
typedef __attribute__((ext_vector_type(8))) float v8f;

#define KDIM 64          // num embeddings (rows of weight / table)
#define DDIM 64          // embedding dim
#define NBINS 63
#define ROWS_PER_BLOCK 256

// ---------------------------------------------------------------------------
// Kernel 1 (single block): T = S @ W via V_WMMA_F32_16X16X4_F32, T -> d_ws
//   S[i][k] = 1/(|i-k|+1)  (exact f32 reciprocals from a precomputed table)
// ---------------------------------------------------------------------------
__global__ __launch_bounds__(256)
void emb_table_kernel(const float* __restrict__ weight,  // [64*64]
                      float* __restrict__ T) {           // [64*64] out (d_ws)
    __shared__ float s_R[KDIM];          // s_R[d] = 1/(d+1), exact RNE division

    const int tid  = threadIdx.x;
    const int lane = tid & 31;
    const int wave = tid >> 5;

    if (tid < KDIM) s_R[tid] = 1.0f / (float)(tid + 1);   // 64 exact divisions, once per grid
    __syncthreads();

    // A (16x4 f32): lanes 0-15 -> K = kb+{0,1} (vgpr .x,.y); lanes 16-31 -> K = kb+{2,3}
    // B (4x16 f32): same K grouping; lane&15 selects N
    // D (16x16 f32): vgpr v -> row v (lanes 0-15) / row v+8 (lanes 16-31); lane&15 = N
    const int m  = lane & 15;
    const int kh = lane >> 4;
    for (int t = wave; t < 16; t += 8) {
        const int ti = t >> 2;          // tile row (M block)
        const int tj = t & 3;           // tile col (N block)
        const int i  = 16 * ti + m;     // global row of S for A-fragment
        v8f c = {};
        for (int kb = 0; kb < KDIM; kb += 4) {
            const int k0 = kb + 2 * kh;
            int d0 = i - k0;       d0 = d0 < 0 ? -d0 : d0;
            int d1 = i - (k0 + 1); d1 = d1 < 0 ? -d1 : d1;
            v2f a;
            a.x = s_R[d0];                               // S[i][k0]
            a.y = s_R[d1];                               // S[i][k0+1]
            v2f b;
            b.x = weight[(k0    ) * DDIM + 16 * tj + m];
            b.y = weight[(k0 + 1) * DDIM + 16 * tj + m];
            c = __builtin_amdgcn_wmma_f32_16x16x4_f32(
                    /*neg_a=*/false, a, /*neg_b=*/false, b,
                    /*c_mod=*/(short)0, c, /*reuse_a=*/false, /*reuse_b=*/false);
        }
#pragma unroll
        for (int v = 0; v < 8; ++v) {
            T[(16 * ti + v + 8 * kh) * DDIM + 16 * tj + m] = c[v];
        }
    }
}

// ---------------------------------------------------------------------------
// Kernel 2 (2048 blocks): pure-bandwidth gather.
//   idx = #{ j : x > bins[j] };  out[row] = T[idx]  (256 B per row, NT stores)
// ---------------------------------------------------------------------------
__global__ __launch_bounds__(256)
void emb_gather_kernel(const float* __restrict__ x,     // [B*F]
                       const float* __restrict__ low,   // [64]; low[1..63] = bins
                       const v4f* __restrict__ T4g,     // [64*16] v4f (d_ws)
                       v4f* __restrict__ out4,          // [B*F*16] v4f
                       int total_rows) {
    __shared__ __align__(16) v4f s_T4[KDIM * DDIM / 4];  // 16 KB table
    __shared__ float s_bins[NBINS];
    __shared__ int   s_idx[ROWS_PER_BLOCK];

    const int tid = threadIdx.x;

    // stage T into LDS: 1024 x v4f, coalesced from L2
#pragma unroll
    for (int i = 0; i < 4; ++i) s_T4[tid + 256 * i] = T4g[tid + 256 * i];
    if (tid < NBINS) s_bins[tid] = low[tid + 1];
    __syncthreads();

    // bucket index per row (bins sorted -> count of bins < x)
    const int rowBase = blockIdx.x * ROWS_PER_BLOCK;
    {
        const int r = rowBase + tid;
        const float xv = (r < total_rows) ? __builtin_nontemporal_load(x + r) : 0.0f;
        int idx = 0;
#pragma unroll
        for (int j = 0; j < NBINS; ++j) idx += (xv > s_bins[j]) ? 1 : 0;
        s_idx[tid] = idx;
    }
    __syncthreads();

    // 16 threads per row: 64 floats = 16 x v4f = 256 B, coalesced NT stores
    const int sub  = tid & 15;
    const int rsub = tid >> 4;
#pragma unroll 4
    for (int it = 0; it < 16; ++it) {
        const int rl = rsub + 16 * it;
        const int r  = rowBase + rl;
        if (r < total_rows) {
            const int idx = s_idx[rl];
            const v4f val = s_T4[idx * 16 + sub];
            __builtin_nontemporal_store(val, out4 + (long long)r * 16 + sub);
        }
    }
}

extern "C" void kernel_launch(void* const* d_in, const int* in_sizes, int n_in,
                              void* d_out, int out_size, void* d_ws, size_t ws_size,
                              hipStream_t stream) {
    const float* x      = (const float*)d_in[0];   // [8192*64]
    const float* low    = (const float*)d_in[1];   // [64]
    // d_in[2] = high (unused: bins sorted -> index = count of bins < x)
    const float* weight = (const float*)d_in[3];   // [64*64]

    float* T = (float*)d_ws;                       // 16 KB table scratch

    emb_table_kernel<<<1, 256, 0, stream>>>(weight, T);

    const int total_rows = in_sizes[0];            // B*F = 524288
    const int blocks = (total_rows + ROWS_PER_BLOCK - 1) / ROWS_PER_BLOCK;
    emb_gather_kernel<<<blocks, 256, 0, stream>>>(
        x, low, (const v4f*)T, (v4f*)d_out, total_rows);
}